// EvolveGCN_7189775253563
// MI455X (gfx1250) — compile-verified
//
#include <hip/hip_runtime.h>
#include <cstdint>
#include <cstddef>

#define T_STEPS 6
#define N_NODES 50000
#define N_PAD 50048  // N rounded up to 64-row GEMM tiles; xw is padded workspace
#define N_EDGES 1600000
#define D 128
#define RRELU_SLOPE 0.22916666666666666f

// LDS geometry for the WMMA GEMM
#define SZ_LD 132      // padded Z-tile row (words): lane bank stride 4, conflict-free
#define QP_STRIDE 288  // padded packed-Q pair-row (words): 256+32 so khalf=2 lanes
                       // (one pair-row further) land on banks 32..63 vs 0..31

typedef __attribute__((ext_vector_type(2))) float v2f;
typedef __attribute__((ext_vector_type(8))) float v8f;

// ---------------------------------------------------------------------------
// order-preserving float -> uint key (for top-k threshold search)
// ---------------------------------------------------------------------------
__device__ __forceinline__ unsigned fkey(float f) {
  unsigned u = __float_as_uint(f);
  return (u & 0x80000000u) ? ~u : (u | 0x80000000u);
}

// ---------------------------------------------------------------------------
// ||scorer||^-1  (one block, 128 threads)
// ---------------------------------------------------------------------------
__global__ __launch_bounds__(128) void norm_kernel(const float* __restrict__ s,
                                                   float* __restrict__ invn) {
  __shared__ float red[128];
  int tid = threadIdx.x;
  float v = s[tid];
  red[tid] = v * v;
  __syncthreads();
  for (int o = 64; o > 0; o >>= 1) {
    if (tid < o) red[tid] += red[tid + o];
    __syncthreads();
  }
  if (tid == 0) invn[0] = rsqrtf(red[0]);
}

// ---------------------------------------------------------------------------
// scores[i] = dot(Z[i,:], scorer) * invnorm   (one wave32 per node;
// grid exactly tiles N/8 -> no bounds guard needed)
// ---------------------------------------------------------------------------
__global__ __launch_bounds__(256) void score_kernel(const float* __restrict__ Z,
                                                    const float* __restrict__ scorer,
                                                    const float* __restrict__ invn,
                                                    float* __restrict__ scores) {
  const int lane = threadIdx.x & 31;
  const int wv = threadIdx.x >> 5;
  const int node = blockIdx.x * 8 + wv;
  float4 z = ((const float4*)(Z + (size_t)node * D))[lane];
  float4 s = ((const float4*)scorer)[lane];
  float d = z.x * s.x + z.y * s.y + z.z * s.z + z.w * s.w;
  for (int o = 16; o > 0; o >>= 1) d += __shfl_xor(d, o, 32);
  if (lane == 0) scores[node] = d * invn[0];
}

// ---------------------------------------------------------------------------
// top-128 of scores: binary search on uint keys + tie collection + bitonic
// sort (desc by value, asc by index) to reproduce jax.lax.top_k rank order.
// One workgroup, 1024 threads. Emits selIdx[128], selScale[128]=tanh(val).
// ---------------------------------------------------------------------------
__global__ __launch_bounds__(1024) void topk_kernel(const float* __restrict__ scores, int n,
                                                    int* __restrict__ selIdx,
                                                    float* __restrict__ selScale) {
  __shared__ unsigned s_cnt;
  __shared__ int s_cgt, s_ceq;
  __shared__ unsigned kkey[128];
  __shared__ int kidx[128];
  __shared__ int eqidx[128];
  const int tid = threadIdx.x;

  unsigned lo = 0u, hi = 0xFFFFFFFFu;
  for (int it = 0; it < 33; ++it) {
    if (lo >= hi) break;
    unsigned mid = (unsigned)(((unsigned long long)lo + (unsigned long long)hi + 1ull) >> 1);
    if (tid == 0) s_cnt = 0u;
    __syncthreads();
    unsigned local = 0;
    for (int i = tid; i < n; i += 1024)
      if (fkey(scores[i]) >= mid) ++local;
    atomicAdd(&s_cnt, local);
    __syncthreads();
    unsigned c = s_cnt;
    __syncthreads();
    if (c >= 128u) lo = mid; else hi = mid - 1u;
  }
  const unsigned tau = lo;

  if (tid == 0) { s_cgt = 0; s_ceq = 0; }
  __syncthreads();
  for (int i = tid; i < n; i += 1024) {
    unsigned k = fkey(scores[i]);
    if (k > tau) {
      int p = atomicAdd(&s_cgt, 1);
      if (p < 128) { kkey[p] = k; kidx[p] = i; }
    } else if (k == tau) {
      int q = atomicAdd(&s_ceq, 1);
      if (q < 128) eqidx[q] = i;
    }
  }
  __syncthreads();
  const int cgt = min(s_cgt, 128);
  const int ceq = min(s_ceq, 128);
  if (tid < 128 && tid >= ceq) eqidx[tid] = 0x7FFFFFFF;
  __syncthreads();
  // bitonic ascending sort of tie indices
  for (int k2 = 2; k2 <= 128; k2 <<= 1)
    for (int j = k2 >> 1; j > 0; j >>= 1) {
      if (tid < 128) {
        int ixj = tid ^ j;
        if (ixj > tid) {
          int a = eqidx[tid], b = eqidx[ixj];
          bool doswap = ((tid & k2) == 0) ? (b < a) : (a < b);
          if (doswap) { eqidx[tid] = b; eqidx[ixj] = a; }
        }
      }
      __syncthreads();
    }
  if (tid < 128 && tid >= cgt) {
    int e = tid - cgt;
    int id = (e < ceq) ? eqidx[e] : -1;
    kidx[tid] = id;
    kkey[tid] = (id >= 0) ? tau : 0u;
  }
  __syncthreads();
  // bitonic sort of 128 winners: key desc, idx asc
  for (int k2 = 2; k2 <= 128; k2 <<= 1)
    for (int j = k2 >> 1; j > 0; j >>= 1) {
      if (tid < 128) {
        int ixj = tid ^ j;
        if (ixj > tid) {
          unsigned ka = kkey[tid], kb = kkey[ixj];
          int ia = kidx[tid], ib = kidx[ixj];
          bool bFirst = (kb > ka) || (kb == ka && ib < ia);  // "b should precede a"
          bool doswap = ((tid & k2) == 0) ? bFirst : !bFirst;
          if (doswap) { kkey[tid] = kb; kidx[tid] = ib; kkey[ixj] = ka; kidx[ixj] = ia; }
        }
      }
      __syncthreads();
    }
  if (tid < 128) {
    int id = kidx[tid];
    selIdx[tid] = (id >= 0) ? id : 0;
    float v = (id >= 0) ? scores[id] : 0.f;
    selScale[tid] = tanhf(v);
  }
}

// ---------------------------------------------------------------------------
// matrix-GRU update of Q (128x128), single workgroup, 256 threads.
// LDS (dynamic): zt[16384] (z_topk), srq[16384] (rst then rst*Q), supd[16384].
// ---------------------------------------------------------------------------
__global__ __launch_bounds__(256) void gru_kernel(const float* __restrict__ Z,
                                                  const int* __restrict__ selIdx,
                                                  const float* __restrict__ selScale,
                                                  const float* __restrict__ gW,
                                                  const float* __restrict__ gU,
                                                  const float* __restrict__ gb,
                                                  float* __restrict__ Q) {
  extern __shared__ float smem[];
  float* zt = smem;            // z_topk[f][j]
  float* srq = smem + 16384;   // rst, then rst*Q
  float* supd = smem + 32768;  // upd
  const int tid = threadIdx.x;

  for (int i = tid; i < 16384; i += 256) {
    int f = i >> 7, j = i & 127;
    zt[i] = Z[(size_t)selIdx[j] * D + f] * selScale[j];
  }
  __syncthreads();

  // upd (gate 0) and rst (gate 1)
  for (int m = 0; m < 64; ++m) {
    int o = tid + 256 * m;
    int i = o >> 7, j = o & 127;
    float a0 = gb[o];
    float a1 = gb[16384 + o];
    const float* w0 = gW + i * D;
    const float* u0 = gU + i * D;
    for (int k = 0; k < D; ++k) {
      float zkj = zt[k * D + j];
      float qkj = Q[k * D + j];
      a0 = fmaf(w0[k], zkj, a0);
      a0 = fmaf(u0[k], qkj, a0);
      a1 = fmaf(w0[16384 + k], zkj, a1);
      a1 = fmaf(u0[16384 + k], qkj, a1);
    }
    supd[o] = 1.f / (1.f + __expf(-a0));
    srq[o] = 1.f / (1.f + __expf(-a1));
  }
  __syncthreads();
  for (int i = tid; i < 16384; i += 256) srq[i] *= Q[i];  // rst * Q
  __syncthreads();
  // hcap (gate 2) and final combine
  for (int m = 0; m < 64; ++m) {
    int o = tid + 256 * m;
    int i = o >> 7, j = o & 127;
    float a = gb[32768 + o];
    const float* w2 = gW + 32768 + i * D;
    const float* u2 = gU + 32768 + i * D;
    for (int k = 0; k < D; ++k) {
      a = fmaf(w2[k], zt[k * D + j], a);
      a = fmaf(u2[k], srq[k * D + j], a);
    }
    float h = tanhf(a);
    float u = supd[o];
    Q[o] = (1.f - u) * Q[o] + u * h;
  }
}

// ---------------------------------------------------------------------------
// xw = Z @ Q  via V_WMMA_F32_16X16X4_F32 (fp32 in/out, matches reference).
// Block tile: 64 rows x 128 cols, 8 waves. xw has N_PAD rows so the epilogue
// stores are unguarded (padded rows compute to zero from zero-filled sZ).
// sZ: 64x132 padded Z tile (A fragments: one aligned ds_load_b64, bank-clean).
// sQp: Q pre-packed in B-fragment order — pair-row p holds interleaved
//      {Q[2p][n], Q[2p+1][n]} so every B fragment is one aligned ds_load_b64;
//      288-word pair-row pad puts lane groups 0-15 / 16-31 on disjoint banks.
// ---------------------------------------------------------------------------
__global__ __launch_bounds__(256) void gemm_xw_kernel(const float* __restrict__ Z,
                                                      const float* __restrict__ Q,
                                                      float* __restrict__ xw, int n) {
  extern __shared__ float smem[];
  float* sZ = smem;               // 64 x SZ_LD
  float* sQp = smem + 64 * SZ_LD; // 64 pair-rows x QP_STRIDE
  const int tid = threadIdx.x;
  const int rowBase = blockIdx.x * 64;

  // pack Q into fragment order
  for (int i = tid; i < D * (D / 4); i += 256) {
    int k = i >> 5, c4 = (i & 31) * 4;
    float4 v = ((const float4*)(Q + (size_t)k * D))[i & 31];
    float* dst = sQp + (k >> 1) * QP_STRIDE + (k & 1);
    dst[2 * (c4 + 0)] = v.x;
    dst[2 * (c4 + 1)] = v.y;
    dst[2 * (c4 + 2)] = v.z;
    dst[2 * (c4 + 3)] = v.w;
  }
  // load Z tile (guarded: real input is only n rows)
  for (int i = tid; i < 64 * (D / 4); i += 256) {
    int r = i >> 5, c4 = i & 31;
    int gr = rowBase + r;
    float4 v = make_float4(0.f, 0.f, 0.f, 0.f);
    if (gr < n) v = ((const float4*)(Z + (size_t)gr * D))[c4];
    *(float4*)(sZ + r * SZ_LD + c4 * 4) = v;
  }
  __syncthreads();

  const int wave = tid >> 5, lane = tid & 31;
  const int m0 = (wave >> 1) * 16;  // 0,16,32,48
  const int n0 = (wave & 1) * 64;   // 0 or 64
  const int lrow = lane & 15;
  const int khalf = (lane >> 4) * 2;  // A/B K-half per ISA 16x4 layout
  const float* arow = sZ + (m0 + lrow) * SZ_LD + khalf;
  v8f acc[4] = {};

  for (int k0 = 0; k0 < D; k0 += 4) {
    v2f a = *(const v2f*)(arow + k0);
    const float* bbase = sQp + ((k0 + khalf) >> 1) * QP_STRIDE;
#pragma unroll
    for (int nt = 0; nt < 4; ++nt) {
      v2f b = *(const v2f*)(bbase + 2 * (n0 + nt * 16 + lrow));
      acc[nt] = __builtin_amdgcn_wmma_f32_16x16x4_f32(false, a, false, b, (short)0,
                                                      acc[nt], false, false);
    }
  }

  // C layout: VGPR r -> M = r + 8*(lane/16), N = lane%16. Unguarded stores
  // into the padded xw buffer: no per-store exec-mask branching.
  float* obase = xw + (size_t)(rowBase + m0 + 8 * (lane >> 4)) * D + n0 + (lane & 15);
#pragma unroll
  for (int nt = 0; nt < 4; ++nt)
#pragma unroll
    for (int r = 0; r < 8; ++r)
      obase[(size_t)r * D + nt * 16] = acc[nt][r];
}

// ---------------------------------------------------------------------------
// ax = 0  (grid exactly tiles n4)
// ---------------------------------------------------------------------------
__global__ __launch_bounds__(256) void zero_kernel(float4* __restrict__ p) {
  int i = blockIdx.x * 256 + threadIdx.x;
  p[i] = make_float4(0.f, 0.f, 0.f, 0.f);
}

// ---------------------------------------------------------------------------
// ax[dst,:] += w * xw[src,:]  — one wave32 per edge, float4 slice per lane,
// L2-resident atomics (xw/ax are 25.6 MB each; fit MI455X's 192 MB L2).
// Grid exactly tiles E (E % 8 == 0).
// ---------------------------------------------------------------------------
__global__ __launch_bounds__(256) void edge_kernel(const int* __restrict__ src,
                                                   const int* __restrict__ dst,
                                                   const float* __restrict__ w,
                                                   const float* __restrict__ xw,
                                                   float* __restrict__ ax) {
  const int lane = threadIdx.x & 31;
  const int wv = threadIdx.x >> 5;
  const long long e = (long long)blockIdx.x * 8 + wv;
  const int s = src[e];
  const int d = dst[e];
  const float ww = w[e];
  float4 v = ((const float4*)(xw + (size_t)s * D))[lane];
  float* base = ax + (size_t)d * D + lane * 4;
  atomicAdd(base + 0, ww * v.x);
  atomicAdd(base + 1, ww * v.y);
  atomicAdd(base + 2, ww * v.z);
  atomicAdd(base + 3, ww * v.w);
}

// ---------------------------------------------------------------------------
// out = rrelu(ax) (eval-mode slope; grid exactly tiles n4)
// ---------------------------------------------------------------------------
__global__ __launch_bounds__(256) void rrelu_kernel(const float4* __restrict__ in,
                                                    float4* __restrict__ out_) {
  int i = blockIdx.x * 256 + threadIdx.x;
  float4 v = in[i];
  v.x = v.x >= 0.f ? v.x : RRELU_SLOPE * v.x;
  v.y = v.y >= 0.f ? v.y : RRELU_SLOPE * v.y;
  v.z = v.z >= 0.f ? v.z : RRELU_SLOPE * v.z;
  v.w = v.w >= 0.f ? v.w : RRELU_SLOPE * v.w;
  out_[i] = v;
}

// ---------------------------------------------------------------------------
// host orchestration
// ---------------------------------------------------------------------------
extern "C" void kernel_launch(void* const* d_in, const int* in_sizes, int n_in,
                              void* d_out, int out_size, void* d_ws, size_t ws_size,
                              hipStream_t stream) {
  (void)in_sizes; (void)n_in; (void)out_size; (void)ws_size;
  const float* nodes  = (const float*)d_in[0];
  const int* esrc     = (const int*)d_in[1];
  const int* edst     = (const int*)d_in[2];
  const float* ew     = (const float*)d_in[3];
  const float* Winit1 = (const float*)d_in[4];
  const float* scor1  = (const float*)d_in[5];
  const float* gW1    = (const float*)d_in[6];
  const float* gU1    = (const float*)d_in[7];
  const float* gb1    = (const float*)d_in[8];
  const float* Winit2 = (const float*)d_in[9];
  const float* scor2  = (const float*)d_in[10];
  const float* gW2    = (const float*)d_in[11];
  const float* gU2    = (const float*)d_in[12];
  const float* gb2    = (const float*)d_in[13];
  float* out = (float*)d_out;

  char* ws = (char*)d_ws;
  size_t off = 0;
  auto take = [&](size_t bytes) -> char* {
    char* p = ws + off;
    off = (off + bytes + 255) & ~(size_t)255;
    return p;
  };
  float* Q1       = (float*)take((size_t)D * D * 4);
  float* Q2       = (float*)take((size_t)D * D * 4);
  float* inv1     = (float*)take(4);
  float* inv2     = (float*)take(4);
  int* selIdx     = (int*)take((size_t)D * 4);
  float* selScale = (float*)take((size_t)D * 4);
  float* scores   = (float*)take((size_t)N_NODES * 4);
  float* hbuf     = (float*)take((size_t)N_NODES * D * 4);
  float* xw       = (float*)take((size_t)N_PAD * D * 4);  // padded rows
  float* ax       = (float*)take((size_t)N_NODES * D * 4);

  hipMemcpyAsync(Q1, Winit1, (size_t)D * D * 4, hipMemcpyDeviceToDevice, stream);
  hipMemcpyAsync(Q2, Winit2, (size_t)D * D * 4, hipMemcpyDeviceToDevice, stream);
  norm_kernel<<<1, 128, 0, stream>>>(scor1, inv1);
  norm_kernel<<<1, 128, 0, stream>>>(scor2, inv2);

  const int n4 = N_NODES * D / 4;
  const int scoreBlocks = N_NODES / 8;           // exact
  const int gemmBlocks = N_PAD / 64;             // exact (782)
  const size_t gemmShmem = (size_t)(64 * SZ_LD + 64 * QP_STRIDE) * 4;
  const size_t gruShmem = (size_t)3 * D * D * 4;
  const int edgeBlocks = N_EDGES / 8;            // exact
  const int ewBlocks = n4 / 256;                 // exact

  for (int t = 0; t < T_STEPS; ++t) {
    const float* Zt = nodes + (size_t)t * N_NODES * D;
    const int* st = esrc + (size_t)t * N_EDGES;
    const int* dt = edst + (size_t)t * N_EDGES;
    const float* wt = ew + (size_t)t * N_EDGES;
    for (int layer = 0; layer < 2; ++layer) {
      const float* Z      = (layer == 0) ? Zt : hbuf;
      const float* scorer = (layer == 0) ? scor1 : scor2;
      const float* inv    = (layer == 0) ? inv1 : inv2;
      const float* gW     = (layer == 0) ? gW1 : gW2;
      const float* gU     = (layer == 0) ? gU1 : gU2;
      const float* gb     = (layer == 0) ? gb1 : gb2;
      float* Q            = (layer == 0) ? Q1 : Q2;
      float* dest         = (layer == 0) ? hbuf : out + (size_t)t * N_NODES * D;

      score_kernel<<<scoreBlocks, 256, 0, stream>>>(Z, scorer, inv, scores);
      topk_kernel<<<1, 1024, 0, stream>>>(scores, N_NODES, selIdx, selScale);
      gru_kernel<<<1, 256, gruShmem, stream>>>(Z, selIdx, selScale, gW, gU, gb, Q);
      gemm_xw_kernel<<<gemmBlocks, 256, gemmShmem, stream>>>(Z, Q, xw, N_NODES);
      zero_kernel<<<ewBlocks, 256, 0, stream>>>((float4*)ax);
      edge_kernel<<<edgeBlocks, 256, 0, stream>>>(st, dt, wt, xw, ax);
      rrelu_kernel<<<ewBlocks, 256, 0, stream>>>((const float4*)ax, (float4*)dest);
    }
  }
}